// RHOFM_30648886624574
// MI455X (gfx1250) — compile-verified
//
#include <hip/hip_runtime.h>

// RHOFM order-3 via Newton's identities:
//   p_k = (x^k) @ (w^k)  (three f32 WMMA GEMMs, K=256, N=32)
//   e2 = (p1^2 - p2)/2 ; e3 = (p1^3 - 3 p1 p2 + 2 p3)/6
//   out = th0 + sum_d( p1*th1[d] + th2[0]*e2 + th2[1]*e3 )

typedef __attribute__((ext_vector_type(2))) float v2f;
typedef __attribute__((ext_vector_type(8))) float v8f;

#define FDIM 128
#define DDIM 32
#define ROWS_PER_WAVE 16
#define WAVES_PER_BLOCK 8
#define THREADS (WAVES_PER_BLOCK * 32)

__global__ __launch_bounds__(THREADS) void rhofm_wmma_kernel(
    const float* __restrict__ item,   // (n, 128)
    const float* __restrict__ user,   // (n, 128)
    const float* __restrict__ embs,   // (128, 32)
    const float* __restrict__ theta0, // (1,)
    const float* __restrict__ theta1, // (32,)
    const float* __restrict__ theta2, // (2,)
    float* __restrict__ out,          // (n,)
    int n)
{
    __shared__ float lds_w[FDIM * DDIM];  // 16 KB staged copy of embs

    const int tid = threadIdx.x;
    // Cooperative, fully coalesced stage of embs into LDS.
    for (int t = tid; t < FDIM * DDIM; t += THREADS) lds_w[t] = embs[t];
    __syncthreads();

    const int lane = tid & 31;
    const int wave = tid >> 5;
    const int half = lane >> 4;   // 0: lanes 0-15, 1: lanes 16-31
    const int r    = lane & 15;   // M row (A) / N col (B/C) within half
    const int m0   = (blockIdx.x * WAVES_PER_BLOCK + wave) * ROWS_PER_WAVE;
    const int row  = m0 + r;      // this lane's A-matrix row

    // acc[p][t]: p = power-1 (p1,p2,p3), t = N-tile (cols 0-15 / 16-31)
    v8f acc[3][2] = {};

    // Feature loop, split into item-half and user-half (w repeats embs).
    for (int src = 0; src < 2; ++src) {
        const float* __restrict__ rowp =
            (src == 0 ? item : user) + (size_t)row * FDIM;
        for (int k0 = 0; k0 < FDIM; k0 += 4) {
            // A 16x4 f32 layout: VGPR0 = {K0 | K2}, VGPR1 = {K1 | K3}
            // across lane halves -> lane loads features ka, ka+1 of its row.
            const int ka = k0 + 2 * half;
            v2f a1;
            a1.x = rowp[ka + 0];
            a1.y = rowp[ka + 1];
            v2f a2 = a1 * a1;
            v2f a3 = a2 * a1;

            // B 4x16 f32: rows striped across lanes; same K split as A.
            v2f b1_0, b1_1;
            b1_0.x = lds_w[(ka + 0) * DDIM + r];
            b1_0.y = lds_w[(ka + 1) * DDIM + r];
            b1_1.x = lds_w[(ka + 0) * DDIM + 16 + r];
            b1_1.y = lds_w[(ka + 1) * DDIM + 16 + r];
            v2f b2_0 = b1_0 * b1_0, b3_0 = b2_0 * b1_0;
            v2f b2_1 = b1_1 * b1_1, b3_1 = b2_1 * b1_1;

            // 6 independent accumulator chains -> good WMMA pipelining.
            acc[0][0] = __builtin_amdgcn_wmma_f32_16x16x4_f32(
                false, a1, false, b1_0, (short)0, acc[0][0], false, false);
            acc[0][1] = __builtin_amdgcn_wmma_f32_16x16x4_f32(
                false, a1, false, b1_1, (short)0, acc[0][1], false, false);
            acc[1][0] = __builtin_amdgcn_wmma_f32_16x16x4_f32(
                false, a2, false, b2_0, (short)0, acc[1][0], false, false);
            acc[1][1] = __builtin_amdgcn_wmma_f32_16x16x4_f32(
                false, a2, false, b2_1, (short)0, acc[1][1], false, false);
            acc[2][0] = __builtin_amdgcn_wmma_f32_16x16x4_f32(
                false, a3, false, b3_0, (short)0, acc[2][0], false, false);
            acc[2][1] = __builtin_amdgcn_wmma_f32_16x16x4_f32(
                false, a3, false, b3_1, (short)0, acc[2][1], false, false);
        }
    }

    // Epilogue: Newton's identities + N-reduction.
    const float t0  = theta0[0];
    const float t20 = theta2[0];
    const float t21 = theta2[1];
    const float th1a = theta1[r];       // column of N-tile 0 this lane holds
    const float th1b = theta1[16 + r];  // column of N-tile 1

    float rowsum[8];
#pragma unroll
    for (int i = 0; i < 8; ++i) {
        float v = 0.0f;
        {
            float p1 = acc[0][0][i], p2 = acc[1][0][i], p3 = acc[2][0][i];
            float e2 = 0.5f * (p1 * p1 - p2);
            float e3 = (e2 * p1 - p1 * p2 + p3) * (1.0f / 3.0f);
            v += th1a * p1 + t20 * e2 + t21 * e3;
        }
        {
            float p1 = acc[0][1][i], p2 = acc[1][1][i], p3 = acc[2][1][i];
            float e2 = 0.5f * (p1 * p1 - p2);
            float e3 = (e2 * p1 - p1 * p2 + p3) * (1.0f / 3.0f);
            v += th1b * p1 + t20 * e2 + t21 * e3;
        }
        // Sum over the 16 lanes of this half (N = 0..15 of both tiles).
        // C layout: lanes 0-15 hold M=i, lanes 16-31 hold M=8+i, so xor
        // masks 1,2,4,8 stay within a half on wave32.
        v += __shfl_xor(v, 1, 32);
        v += __shfl_xor(v, 2, 32);
        v += __shfl_xor(v, 4, 32);
        v += __shfl_xor(v, 8, 32);
        rowsum[i] = v;
    }

    if (r == 0) {
        const int mbase = m0 + half * 8;  // lane 0 -> rows 0-7, lane 16 -> rows 8-15
#pragma unroll
        for (int i = 0; i < 8; ++i) {
            if (mbase + i < n) out[mbase + i] = t0 + rowsum[i];
        }
    }
}

extern "C" void kernel_launch(void* const* d_in, const int* in_sizes, int n_in,
                              void* d_out, int out_size, void* d_ws, size_t ws_size,
                              hipStream_t stream) {
    (void)n_in; (void)d_ws; (void)ws_size; (void)out_size;
    const float* item = (const float*)d_in[0];
    const float* user = (const float*)d_in[1];
    const float* embs = (const float*)d_in[2];
    const float* th0  = (const float*)d_in[3];
    const float* th1  = (const float*)d_in[4];
    const float* th2  = (const float*)d_in[5];
    float* out = (float*)d_out;

    const int n = in_sizes[0] / FDIM;                       // 8192
    const int rows_per_block = ROWS_PER_WAVE * WAVES_PER_BLOCK;  // 128
    const int grid = (n + rows_per_block - 1) / rows_per_block;  // 64

    rhofm_wmma_kernel<<<grid, THREADS, 0, stream>>>(item, user, embs,
                                                    th0, th1, th2, out, n);
}